// FGN_layer_66262755443328
// MI455X (gfx1250) — compile-verified
//
#include <hip/hip_runtime.h>

typedef __attribute__((ext_vector_type(2))) float v2f;
typedef __attribute__((ext_vector_type(8))) float v8f;

#define B_DIM 1024
#define I_DIM 512
#define O_DIM 512
#define EPSI 1e-32f

// ---------------------------------------------------------------------------
// Kernel 1: Kc[o] = sum_i centers[o,i]^2 * (inv_covars[o,i]^2 + EPSI)
// ---------------------------------------------------------------------------
__global__ __launch_bounds__(128)
void fgn_kconst(const float* __restrict__ centers,
                const float* __restrict__ invc,
                float* __restrict__ Kc) {
  __shared__ float red[128];
  const int o = blockIdx.x;
  const int t = threadIdx.x;
  const float* c = centers + (size_t)o * I_DIM;
  const float* s = invc    + (size_t)o * I_DIM;
  float acc = 0.0f;
  for (int i = t; i < I_DIM; i += 128) {
    const float cv = c[i];
    const float sv = s[i];
    const float s2 = sv * sv + EPSI;
    acc += cv * cv * s2;
  }
  red[t] = acc;
  __syncthreads();
  for (int w = 64; w > 0; w >>= 1) {
    if (t < w) red[t] += red[t + w];
    __syncthreads();
  }
  if (t == 0) Kc[o] = red[0];
}

// ---------------------------------------------------------------------------
// Async global -> LDS tile copy, 16B per lane, GVS addressing
// (vdst = per-lane LDS byte address, vaddr = per-lane 32-bit byte offset,
//  saddr = uniform 64-bit base). Tracked with ASYNCcnt.
// ---------------------------------------------------------------------------
__device__ __forceinline__
void async_b128(unsigned lds_addr, const float* __restrict__ base, unsigned byte_off) {
  asm volatile("global_load_async_to_lds_b128 %0, %1, %2"
               :: "v"(lds_addr), "v"(byte_off), "s"(base)
               : "memory");
}

// ---------------------------------------------------------------------------
// Kernel 2: fused triple-GEMM + epilogue
//   L  = X . W^T ; G1 = X^2 . S2^T ; G2 = X . (C*S2)^T  (S2 = ic^2 + eps)
//   out = (L + bias) * exp(-(G1 - 2*G2 + Kc))
// 256 thr = 8 waves; block tile 64x64; wave grid 4(M) x 2(N); K slabs of 16
// double-buffered in LDS via GLOBAL_LOAD_ASYNC_TO_LDS_B128.
// ---------------------------------------------------------------------------
__global__ __launch_bounds__(256)
void fgn_main(const float* __restrict__ X,
              const float* __restrict__ W,
              const float* __restrict__ bias,
              const float* __restrict__ C,
              const float* __restrict__ S,
              const float* __restrict__ Kc,
              float* __restrict__ out) {
  constexpr int KT = 16;            // K slab per stage
  constexpr int LD = 20;            // 80B row stride: 16B aligned, conflict-free
  constexpr int NSLAB = I_DIM / KT; // 32
  // [buffer][operand: 0=X 1=W 2=ic 3=c][row][col]
  __shared__ float lds[2][4][64][LD];

  const int mBase = blockIdx.y * 64;
  const int nBase = blockIdx.x * 64;
  const int t     = threadIdx.x;
  const int lane  = t & 31;
  const int wave  = t >> 5;
  const int waveM = wave & 3;                 // 0..3 : 16-row tile
  const int waveN = wave >> 2;                // 0..1 : 32-col strip
  const int lrow  = lane & 15;
  const int kHalf = (lane < 16) ? 0 : 2;      // ISA 16x16x4 lane layout

  const int ldRow = t >> 2;                   // 0..63
  const int ldCol = (t & 3) * 4;              // 0,4,8,12

  // per-lane global element offsets (col k added per slab)
  const unsigned rowEX = (unsigned)((mBase + ldRow) * I_DIM + ldCol);
  const unsigned rowEO = (unsigned)((nBase + ldRow) * I_DIM + ldCol);

  v8f accL[2]  = {v8f{}, v8f{}};
  v8f accG1[2] = {v8f{}, v8f{}};
  v8f accG2[2] = {v8f{}, v8f{}};

  auto issue = [&](int slab, int buf) {
    const unsigned k0 = (unsigned)slab * KT;
    async_b128((unsigned)(size_t)&lds[buf][0][ldRow][ldCol], X, (rowEX + k0) * 4u);
    async_b128((unsigned)(size_t)&lds[buf][1][ldRow][ldCol], W, (rowEO + k0) * 4u);
    async_b128((unsigned)(size_t)&lds[buf][2][ldRow][ldCol], S, (rowEO + k0) * 4u);
    async_b128((unsigned)(size_t)&lds[buf][3][ldRow][ldCol], C, (rowEO + k0) * 4u);
  };

  issue(0, 0);   // prime the pipeline

  for (int s = 0; s < NSLAB; ++s) {
    const int buf = s & 1;
    if (s + 1 < NSLAB) {
      issue(s + 1, buf ^ 1);                           // overlap next slab copy
      asm volatile("s_wait_asynccnt 0x4" ::: "memory"); // slab s (oldest 4) done
    } else {
      asm volatile("s_wait_asynccnt 0x0" ::: "memory");
    }
    __syncthreads();   // all waves' slab-s tiles visible in LDS

    const float (*As)[LD]  = lds[buf][0];
    const float (*Bw)[LD]  = lds[buf][1];
    const float (*Bic)[LD] = lds[buf][2];
    const float (*Bc)[LD]  = lds[buf][3];

    #pragma unroll
    for (int kk = 0; kk < KT; kk += 4) {
      const int kc = kk + kHalf;
      v2f a, a2;
      {
        const float2 av = *(const float2*)&As[waveM * 16 + lrow][kc];
        a[0] = av.x;         a[1] = av.y;
        a2[0] = av.x * av.x; a2[1] = av.y * av.y;   // X^2 fragment is free
      }
      #pragma unroll
      for (int nt = 0; nt < 2; ++nt) {
        const int nr = waveN * 32 + nt * 16 + lrow;
        v2f bw, b2, bc;
        {
          const float2 vw = *(const float2*)&Bw [nr][kc];
          const float2 vi = *(const float2*)&Bic[nr][kc];
          const float2 vc = *(const float2*)&Bc [nr][kc];
          bw[0] = vw.x;                      bw[1] = vw.y;
          b2[0] = fmaf(vi.x, vi.x, EPSI);    b2[1] = fmaf(vi.y, vi.y, EPSI);
          bc[0] = vc.x * b2[0];              bc[1] = vc.y * b2[1];
        }
        accL[nt]  = __builtin_amdgcn_wmma_f32_16x16x4_f32(
            false, a,  false, bw, (short)0, accL[nt],  false, false);
        accG1[nt] = __builtin_amdgcn_wmma_f32_16x16x4_f32(
            false, a2, false, b2, (short)0, accG1[nt], false, false);
        accG2[nt] = __builtin_amdgcn_wmma_f32_16x16x4_f32(
            false, a,  false, bc, (short)0, accG2[nt], false, false);
      }
    }
    __syncthreads();   // slab consumed; its buffer may be overwritten next iter
  }

  // ---- epilogue: out = (L + bias) * exp(-(G1 - 2*G2 + Kc)) ----
  #pragma unroll
  for (int nt = 0; nt < 2; ++nt) {
    const int o    = nBase + waveN * 32 + nt * 16 + lrow;
    const float bo = bias[o];
    const float ko = Kc[o];
    const int mTop = mBase + waveM * 16 + ((lane < 16) ? 0 : 8);
    #pragma unroll
    for (int r = 0; r < 8; ++r) {
      const float l = accL[nt][r] + bo;
      const float g = accG1[nt][r] - 2.0f * accG2[nt][r] + ko;
      out[(size_t)(mTop + r) * O_DIM + o] = l * __expf(-g);
    }
  }
}

// ---------------------------------------------------------------------------
extern "C" void kernel_launch(void* const* d_in, const int* in_sizes, int n_in,
                              void* d_out, int out_size, void* d_ws, size_t ws_size,
                              hipStream_t stream) {
  const float* X    = (const float*)d_in[0];   // inputs     [B, I]
  const float* W    = (const float*)d_in[1];   // weights    [O, I]
  const float* bias = (const float*)d_in[2];   // biases     [O]
  const float* C    = (const float*)d_in[3];   // centers    [O, I]
  const float* S    = (const float*)d_in[4];   // inv_covars [O, I]
  float* out = (float*)d_out;
  float* Kc  = (float*)d_ws;                   // 512 floats of scratch

  fgn_kconst<<<O_DIM, 128, 0, stream>>>(C, S, Kc);

  dim3 grid(O_DIM / 64, B_DIM / 64);           // 8 x 16 = 128 blocks
  fgn_main<<<grid, 256, 0, stream>>>(X, W, bias, C, S, Kc, out);
}